// GraphEncoder_27204322853289
// MI455X (gfx1250) — compile-verified
//
#include <hip/hip_runtime.h>
#include <hip/hip_bf16.h>

typedef float v2f __attribute__((ext_vector_type(2)));
typedef float v8f __attribute__((ext_vector_type(8)));

#define DCH 32

__device__ __forceinline__ void atomAddF(float* p, float v) {
    __hip_atomic_fetch_add(p, v, __ATOMIC_RELAXED, __HIP_MEMORY_SCOPE_AGENT);
}

// ---------------- elementwise helpers ----------------
__global__ void k_fill(float* __restrict__ p, float v, int n) {
    int i = blockIdx.x * blockDim.x + threadIdx.x;
    if (i < n) p[i] = v;
}

__global__ void k_deg_scatter(const int* __restrict__ ei, float* __restrict__ deg, int E) {
    int e = blockIdx.x * blockDim.x + threadIdx.x;
    if (e < E) atomAddF(&deg[ei[E + e]], 1.0f);
}

__global__ void k_rsqrt(float* __restrict__ p, int n) {
    int i = blockIdx.x * blockDim.x + threadIdx.x;
    if (i < n) p[i] = rsqrtf(p[i]);
}

// h[i,c] = b[c] + sum_k x[i,k] * w[k,c]   (F_IN = 6)
__global__ void k_lin_in(const float* __restrict__ x, const float* __restrict__ w,
                         const float* __restrict__ b, float* __restrict__ h, int N) {
    int t = blockIdx.x * blockDim.x + threadIdx.x;
    if (t >= N * DCH) return;
    int i = t >> 5, c = t & 31;
    const float* xi = x + (size_t)i * 6;
    float acc = b[c];
#pragma unroll
    for (int k = 0; k < 6; ++k) acc += xi[k] * w[k * DCH + c];
    h[t] = acc;
}

// ---------------- WMMA f32 GEMM: O[n,0:32] = A[n,0:32] @ W[32,32] (+bias)(+relu)
// One wave per 16-row block (grid-stride). B fragments (weights) preloaded once
// per wave into registers and reused across all row blocks.
// Fragment layouts per CDNA5 ISA 7.12.2:
//   A 16x4 f32 : lane m = lane%16, VGPR j holds A[m][k0 + 2*(lane/16) + j]
//   B  4x16 f32: lane n = lane%16, VGPR j holds B[k0 + 2*(lane/16) + j][n]
//   C/D 16x16  : VGPR r -> row r + 8*(lane/16), col lane%16
template <bool RELU>
__global__ void k_wmma_gemm32(const float* __restrict__ A, const float* __restrict__ W,
                              const float* __restrict__ bias, float* __restrict__ O,
                              int nRowBlocks) {
    const int lane = threadIdx.x & 31;
    const int half = lane >> 4;
    const int ln   = lane & 15;
    const int waveId = blockIdx.x * (blockDim.x >> 5) + (threadIdx.x >> 5);
    const int nWaves = gridDim.x * (blockDim.x >> 5);

    // Preload all B fragments: 2 col-tiles x 8 K-steps of 4
    v2f bf[2][8];
#pragma unroll
    for (int t = 0; t < 2; ++t)
#pragma unroll
        for (int s = 0; s < 8; ++s) {
            int k   = 4 * s + 2 * half;
            int col = 16 * t + ln;
            bf[t][s].x = W[k * DCH + col];
            bf[t][s].y = W[(k + 1) * DCH + col];
        }
    float bv[2];
    bv[0] = bias ? bias[ln] : 0.0f;
    bv[1] = bias ? bias[16 + ln] : 0.0f;

    for (int rb = waveId; rb < nRowBlocks; rb += nWaves) {
        const float* arow = A + (size_t)rb * 16 * DCH;
        v2f af[8];
#pragma unroll
        for (int s = 0; s < 8; ++s)
            af[s] = *(const v2f*)(arow + ln * DCH + 4 * s + 2 * half);

        v8f acc[2];
#pragma unroll
        for (int t = 0; t < 2; ++t) {
#pragma unroll
            for (int r = 0; r < 8; ++r) acc[t][r] = bv[t];
#pragma unroll
            for (int s = 0; s < 8; ++s)
                acc[t] = __builtin_amdgcn_wmma_f32_16x16x4_f32(
                    false, af[s], false, bf[t][s], (short)0, acc[t], false, false);
        }

        float* orow = O + (size_t)rb * 16 * DCH;
#pragma unroll
        for (int t = 0; t < 2; ++t)
#pragma unroll
            for (int r = 0; r < 8; ++r) {
                float v = acc[t][r];
                if (RELU) v = fmaxf(v, 0.0f);
                orow[(r + 8 * half) * DCH + 16 * t + ln] = v;
            }
    }
}

// hn[i,c] = conv_b[c] + m[i,c] * dinv[i]^2   (self-loop term + bias, pre-scatter)
__global__ void k_init_hnext(const float* __restrict__ m, const float* __restrict__ dinv,
                             const float* __restrict__ b, float* __restrict__ hn, int N) {
    int t = blockIdx.x * blockDim.x + threadIdx.x;
    if (t >= N * DCH) return;
    int i = t >> 5, c = t & 31;
    float di = dinv[i];
    hn[t] = b[c] + m[t] * di * di;
}

// 4 threads per edge; each does 8 channels: gather m[src]*norm, atomic add into hn[dst]
__global__ void k_edge_scatter(const int* __restrict__ ei, const float* __restrict__ m,
                               const float* __restrict__ dinv, float* __restrict__ hn, int E) {
    long long t = (long long)blockIdx.x * blockDim.x + threadIdx.x;
    if (t >= (long long)E * 4) return;
    int e  = (int)(t >> 2);
    int c0 = ((int)t & 3) * 8;
    int s = ei[e];
    int d = ei[E + e];
    float w = dinv[s] * dinv[d];
    const float4* ms = (const float4*)(m + (size_t)s * DCH + c0);
    float* hd = hn + (size_t)d * DCH + c0;
    float4 v0 = ms[0], v1 = ms[1];
    atomAddF(hd + 0, v0.x * w); atomAddF(hd + 1, v0.y * w);
    atomAddF(hd + 2, v0.z * w); atomAddF(hd + 3, v0.w * w);
    atomAddF(hd + 4, v1.x * w); atomAddF(hd + 5, v1.y * w);
    atomAddF(hd + 6, v1.z * w); atomAddF(hd + 7, v1.w * w);
}

__global__ void k_pool_scatter(const float* __restrict__ h, const int* __restrict__ batch,
                               float* __restrict__ sums, float* __restrict__ cnt, int N) {
    int t = blockIdx.x * blockDim.x + threadIdx.x;
    if (t >= N * DCH) return;
    int i = t >> 5, c = t & 31;
    int g = batch[i];
    atomAddF(&sums[(size_t)g * DCH + c], h[t]);
    if (c == 0) atomAddF(&cnt[g], 1.0f);
}

__global__ void k_pool_final(const float* __restrict__ sums, const float* __restrict__ cnt,
                             float* __restrict__ hg, int G) {
    int t = blockIdx.x * blockDim.x + threadIdx.x;
    if (t >= G * DCH) return;
    int g = t >> 5;
    hg[t] = sums[t] / fmaxf(cnt[g], 1.0f);
}

// prop[g] = p2_b + sum_j t[g,j] * p2_w[j]
__global__ void k_prop(const float* __restrict__ tb, const float* __restrict__ w2,
                       const float* __restrict__ b2, float* __restrict__ out, int G) {
    int g = blockIdx.x * blockDim.x + threadIdx.x;
    if (g >= G) return;
    float acc = b2[0];
    const float* r = tb + (size_t)g * DCH;
#pragma unroll
    for (int j = 0; j < DCH; ++j) acc += r[j] * w2[j];
    out[g] = acc;
}

extern "C" void kernel_launch(void* const* d_in, const int* in_sizes, int n_in,
                              void* d_out, int out_size, void* d_ws, size_t ws_size,
                              hipStream_t stream) {
    const float* x      = (const float*)d_in[0];
    const int*   ei     = (const int*)d_in[1];
    const int*   batch  = (const int*)d_in[2];
    const float* lin_w  = (const float*)d_in[3];
    const float* lin_b  = (const float*)d_in[4];
    const float* conv_w = (const float*)d_in[5];
    const float* conv_b = (const float*)d_in[6];
    const float* mu_w   = (const float*)d_in[7];
    const float* mu_b   = (const float*)d_in[8];
    const float* lv_w   = (const float*)d_in[9];
    const float* lv_b   = (const float*)d_in[10];
    const float* p1_w   = (const float*)d_in[11];
    const float* p1_b   = (const float*)d_in[12];
    const float* p2_w   = (const float*)d_in[13];
    const float* p2_b   = (const float*)d_in[14];
    float* out = (float*)d_out;

    const int N = in_sizes[0] / 6;   // 100000
    const int E = in_sizes[1] / 2;   // 3200000
    const int G = out_size / 65;     // 32 + 32 + 1 outputs per graph -> 2048
    const int L = 4;

    // workspace layout (floats, 256B-aligned chunks)
    float* ws = (float*)d_ws;
    size_t off = 0;
    auto alloc = [&](size_t n) {
        float* p = ws + off;
        off += (n + 63) & ~(size_t)63;
        return p;
    };
    float* dinv = alloc(N);
    float* hA   = alloc((size_t)N * DCH);
    float* hB   = alloc((size_t)N * DCH);
    float* mb   = alloc((size_t)N * DCH);
    float* sums = alloc((size_t)G * (DCH + 1));  // sums[G*32] then cnt[G]
    float* cnt  = sums + (size_t)G * DCH;
    float* hg   = alloc((size_t)G * DCH);
    float* tb   = alloc((size_t)G * DCH);
    if (off * sizeof(float) > ws_size) return;

    const int B = 256;

    // degree (self-loops counted via init=1) -> dinv
    k_fill<<<(N + B - 1) / B, B, 0, stream>>>(dinv, 1.0f, N);
    k_deg_scatter<<<(E + B - 1) / B, B, 0, stream>>>(ei, dinv, E);
    k_rsqrt<<<(N + B - 1) / B, B, 0, stream>>>(dinv, N);

    // input projection
    k_lin_in<<<(N * DCH + B - 1) / B, B, 0, stream>>>(x, lin_w, lin_b, hA, N);

    // conv layers
    float* h  = hA;
    float* hn = hB;
    const int rbN = N / 16;
    const int gemmBlocksN = (rbN + 7) / 8;  // 8 waves per 256-thread block
    for (int l = 0; l < L; ++l) {
        k_wmma_gemm32<false><<<gemmBlocksN, 256, 0, stream>>>(
            h, conv_w + (size_t)l * DCH * DCH, nullptr, mb, rbN);
        k_init_hnext<<<(N * DCH + B - 1) / B, B, 0, stream>>>(
            mb, dinv, conv_b + (size_t)l * DCH, hn, N);
        long long sthreads = (long long)E * 4;
        k_edge_scatter<<<(int)((sthreads + B - 1) / B), B, 0, stream>>>(ei, mb, dinv, hn, E);
        float* tmp = h; h = hn; hn = tmp;
    }

    // global mean pool
    k_fill<<<(G * (DCH + 1) + B - 1) / B, B, 0, stream>>>(sums, 0.0f, G * (DCH + 1));
    k_pool_scatter<<<(N * DCH + B - 1) / B, B, 0, stream>>>(h, batch, sums, cnt, N);
    k_pool_final<<<(G * DCH + B - 1) / B, B, 0, stream>>>(sums, cnt, hg, G);

    // heads: mu, log_var, prop
    const int rbG = G / 16;
    const int gemmBlocksG = (rbG + 7) / 8;
    k_wmma_gemm32<false><<<gemmBlocksG, 256, 0, stream>>>(hg, mu_w, mu_b, out, rbG);
    k_wmma_gemm32<false><<<gemmBlocksG, 256, 0, stream>>>(hg, lv_w, lv_b, out + (size_t)G * DCH, rbG);
    k_wmma_gemm32<true><<<gemmBlocksG, 256, 0, stream>>>(out, p1_w, p1_b, tb, rbG);
    k_prop<<<(G + B - 1) / B, B, 0, stream>>>(tb, p2_w, p2_b, out + (size_t)G * DCH * 2, G);
}